// InfoNCELoss_61735859912993
// MI455X (gfx1250) — compile-verified
//
#include <hip/hip_runtime.h>
#include <hip/hip_bf16.h>

// InfoNCE loss: gather-bound (L2-resident 51MB table), f32 WMMA for the dots.

typedef __attribute__((ext_vector_type(2))) float v2f;
typedef __attribute__((ext_vector_type(4))) float v4f;
typedef __attribute__((ext_vector_type(8))) float v8f;

#define DIM        128
#define NUM_NEG    20
#define BATCH      16384
#define NUM_COLS   (NUM_NEG + 1)     // context + 20 negatives
#define INV_T      (1.0f / 0.07f)
#define WAVES_PER_BLOCK 4
#define TILE_ROWS  16
#define NUM_TILES  (BATCH / TILE_ROWS)            // 1024
#define NUM_BLOCKS (NUM_TILES / WAVES_PER_BLOCK)  // 256

__global__ __launch_bounds__(WAVES_PER_BLOCK * 32)
void infonce_wmma_kernel(const float* __restrict__ emb,
                         const int*   __restrict__ targets,
                         const int*   __restrict__ contexts,
                         const int*   __restrict__ negatives,
                         float*       __restrict__ partials)
{
    // Per-wave LDS: 16x128 f32 target tile (8KB) + 16x16 f32 score tile (1KB)
    __shared__ float ldsT[WAVES_PER_BLOCK][TILE_ROWS * DIM];
    __shared__ float ldsS[WAVES_PER_BLOCK][TILE_ROWS * 16];
    __shared__ float blockRed[WAVES_PER_BLOCK];

    const int wave = threadIdx.x >> 5;
    const int lane = threadIdx.x & 31;
    const int tile = blockIdx.x * WAVES_PER_BLOCK + wave;
    const int base = tile * TILE_ROWS;

    const int row = lane & 15;        // batch row within tile this lane maps to
    const int h   = lane >> 4;        // K-half: lanes 0-15 -> K{0,1}, 16-31 -> K{2,3}

    // ---- Stage target tile into LDS (lane l loads float4 column l of row i) ----
    float* T = &ldsT[wave][0];
    {
        v4f* Tv = (v4f*)T;
        #pragma unroll 4
        for (int i = 0; i < TILE_ROWS; ++i) {
            const int t = targets[base + i];                       // uniform -> s_load
            const v4f* src = (const v4f*)(emb + (size_t)t * DIM);
            Tv[i * (DIM / 4) + lane] = src[lane];
        }
    }
    __syncthreads();

    float m = 0.0f, ssum = 0.0f, pos = 0.0f;

    for (int col = 0; col < NUM_COLS; ++col) {
        // Per-row gather index for this score column
        const int oIdx = (col == 0)
            ? contexts[base + row]
            : negatives[(size_t)(base + row) * NUM_NEG + (col - 1)];
        const float* orow = emb + (size_t)oIdx * DIM;

        // Prefetch next column's row for this lane (global_prefetch_b8)
        if (col < NUM_NEG) {
            const int nIdx = negatives[(size_t)(base + row) * NUM_NEG + col];
            __builtin_prefetch(emb + (size_t)nIdx * DIM, 0, 0);
        }

        // ---- D = T(16x128) x O^T(128x16), f32 WMMA, K accumulated in f32 ----
        v8f acc = {};
        #pragma unroll
        for (int k = 0; k < DIM; k += 4) {
            // 16x4 f32 A fragment: this lane holds A[row][k+2h], A[row][k+2h+1]
            const v2f a = *(const v2f*)&T[row * DIM + k + 2 * h];
            // 4x16 f32 B fragment, mirrored layout: B[kk][n] = orow_n[k+kk]
            const v2f b = *(const v2f*)&orow[k + 2 * h];
            acc = __builtin_amdgcn_wmma_f32_16x16x4_f32(
                /*neg_a=*/false, a, /*neg_b=*/false, b,
                /*c_mod=*/(short)0, acc, /*reuse_a=*/false, /*reuse_b=*/false);
        }

        // ---- Scatter D to LDS per the C/D layout, pick the diagonal ----
        {
            float* S = &ldsS[wave][0];
            const int mb = (lane < 16) ? 0 : 8;
            const int n  = lane & 15;
            #pragma unroll
            for (int g = 0; g < 8; ++g)
                S[(mb + g) * 16 + n] = acc[g];
        }
        __syncthreads();   // uniform across block

        if (lane < 16) {
            const float x = ldsS[wave][row * 16 + row] * INV_T;  // t_row . o_row / T
            if (col == 0) {            // positive score seeds the running logsumexp
                pos = x; m = x; ssum = 1.0f;
            } else if (x > m) {
                ssum = ssum * __expf(m - x) + 1.0f; m = x;
            } else {
                ssum += __expf(x - m);
            }
        }
        __syncthreads();   // score tile reused next column
    }

    // ---- per-row loss = logsumexp - pos ; reduce wave32 -> block -> d_ws ----
    float lr = (lane < 16) ? (m + __logf(ssum) - pos) : 0.0f;
    #pragma unroll
    for (int off = 16; off > 0; off >>= 1)
        lr += __shfl_xor(lr, off, 32);

    if (lane == 0) blockRed[wave] = lr;
    __syncthreads();
    if (threadIdx.x == 0) {
        float s = 0.0f;
        #pragma unroll
        for (int w = 0; w < WAVES_PER_BLOCK; ++w) s += blockRed[w];
        partials[blockIdx.x] = s;
    }
}

// Deterministic fixed-order tree reduction of the 256 block partials.
__global__ __launch_bounds__(256)
void infonce_reduce_kernel(const float* __restrict__ partials,
                           float* __restrict__ out)
{
    __shared__ float s[256];
    const int t = threadIdx.x;
    s[t] = partials[t];
    __syncthreads();
    #pragma unroll
    for (int off = 128; off > 0; off >>= 1) {
        if (t < off) s[t] += s[t + off];
        __syncthreads();
    }
    if (t == 0) out[0] = s[0] * (1.0f / (float)BATCH);
}

extern "C" void kernel_launch(void* const* d_in, const int* in_sizes, int n_in,
                              void* d_out, int out_size, void* d_ws, size_t ws_size,
                              hipStream_t stream) {
    const float* emb       = (const float*)d_in[0];  // [100000,128] f32
    const int*   targets   = (const int*)d_in[1];    // [16384]
    const int*   contexts  = (const int*)d_in[2];    // [16384]
    const int*   negatives = (const int*)d_in[3];    // [16384,20]
    float*       out       = (float*)d_out;          // scalar loss
    float*       partials  = (float*)d_ws;           // 256 floats of scratch

    infonce_wmma_kernel<<<NUM_BLOCKS, WAVES_PER_BLOCK * 32, 0, stream>>>(
        emb, targets, contexts, negatives, partials);
    infonce_reduce_kernel<<<1, 256, 0, stream>>>(partials, out);
}